// NFP_33406255628786
// MI455X (gfx1250) — compile-verified
//
#include <hip/hip_runtime.h>

// Problem constants (from the reference)
#define N_NODES 40000
#define N_EDGES 640000
#define DIM     128     // input feature dim
#define HID     256     // hidden dim
#define NBITS   64      // fingerprint bits
#define DEPTH   2       // loop body is invariant -> compute once, scale by DEPTH

typedef float v2f __attribute__((ext_vector_type(2)));
typedef float v8f __attribute__((ext_vector_type(8)));

// ---------------------------------------------------------------------------
// Kernel 1: zero the h accumulator (N*DIM floats) and the 64-float fps accum.
// ---------------------------------------------------------------------------
__global__ __launch_bounds__(256) void k_zero(float4* __restrict__ h4,
                                              float* __restrict__ fps_acc,
                                              int n4) {
  int i = blockIdx.x * blockDim.x + threadIdx.x;
  if (i < n4) h4[i] = make_float4(0.f, 0.f, 0.f, 0.f);
  if (i < NBITS) fps_acc[i] = 0.f;
}

// ---------------------------------------------------------------------------
// Kernel 2: edge scatter-add  h[dst] += n_feat[src]
// One wave (32 lanes) per edge; lane handles one float4 chunk (32*16B = 512B
// coalesced gather per edge), then 4 global float atomics into h[dst].
// ---------------------------------------------------------------------------
__global__ __launch_bounds__(256) void k_scatter(const float4* __restrict__ nf4,
                                                 const int* __restrict__ src,
                                                 const int* __restrict__ dst,
                                                 float* __restrict__ h) {
  int tid = blockIdx.x * blockDim.x + threadIdx.x;
  int e = tid >> 5;          // edge index
  int c = tid & 31;          // float4 chunk within the 128-float row
  if (e >= N_EDGES) return;
  int s = src[e];
  int d = dst[e];
  float4 v = nf4[(size_t)s * (DIM / 4) + c];
  float* p = h + (size_t)d * DIM + c * 4;
  atomicAdd(p + 0, v.x);
  atomicAdd(p + 1, v.y);
  atomicAdd(p + 2, v.z);
  atomicAdd(p + 3, v.w);
}

// ---------------------------------------------------------------------------
// Kernel 3: fused node MLP per 16-node tile, all matmuls via f32 WMMA.
//   r      = relu(h_tile @ W1 + b1)        [16,256]
//   logits = r @ W2 + b2                   [16,64]
//   probs  = softmax(logits, axis=1)
//   fps_acc[c] += sum_rows(probs[:,c])     (global float atomics, 64/block)
// Block = 128 threads = 4 waves. LDS padded to kill bank conflicts on the
// K-strided A-fragment reads.
// ---------------------------------------------------------------------------
#define HS (DIM + 1)   // padded stride for h tile in LDS
#define RS (HID + 1)   // padded stride for r tile in LDS

__global__ __launch_bounds__(128) void k_node_mlp(const float* __restrict__ h,
                                                  const float* __restrict__ W1,
                                                  const float* __restrict__ b1,
                                                  const float* __restrict__ W2,
                                                  const float* __restrict__ b2,
                                                  float* __restrict__ fps_acc) {
  __shared__ float h_sh[16 * HS];      // 16 x 128 (padded)
  __shared__ float r_sh[16 * RS];      // 16 x 256 (padded)
  __shared__ float l_sh[16 * NBITS];   // 16 x 64 logits -> probs

  const int t    = threadIdx.x;
  const int lane = t & 31;
  const int wv   = t >> 5;        // wave id 0..3
  const int m    = lane & 15;     // matrix row (A/M, or column N for B/C/D)
  const int kh   = lane >> 4;     // 0/1: selects K-pair half (f32 WMMA layout)
  const int nodeBase = blockIdx.x * 16;

  // ---- stage h tile (coalesced) ----
  for (int i = t; i < 16 * DIM; i += 128) {
    int r = i >> 7, c = i & (DIM - 1);
    h_sh[r * HS + c] = h[(size_t)(nodeBase + r) * DIM + c];
  }
  __syncthreads();

  const v8f vzero = {0.f, 0.f, 0.f, 0.f, 0.f, 0.f, 0.f, 0.f};

  // ---- GEMM1: wave wv computes output columns [wv*64, wv*64+64) ----
  v8f acc[4];
  acc[0] = vzero; acc[1] = vzero; acc[2] = vzero; acc[3] = vzero;

  for (int kk = 0; kk < DIM / 4; ++kk) {
    const int kb = kk * 4 + 2 * kh;                 // this lane's K pair base
    v2f a;
    a[0] = h_sh[m * HS + kb];                       // A[m][kb]
    a[1] = h_sh[m * HS + kb + 1];                   // A[m][kb+1]
#pragma unroll
    for (int ct = 0; ct < 4; ++ct) {
      const int col = wv * 64 + ct * 16 + m;
      v2f b;
      b[0] = W1[(size_t)kb * HID + col];            // B[kb][col]
      b[1] = W1[(size_t)(kb + 1) * HID + col];
      acc[ct] = __builtin_amdgcn_wmma_f32_16x16x4_f32(
          false, a, false, b, (short)0, acc[ct], false, false);
    }
  }

  // bias + relu, write r tile to LDS (C/D layout: row = v + 8*kh, col = m)
#pragma unroll
  for (int ct = 0; ct < 4; ++ct) {
    const int col = wv * 64 + ct * 16 + m;
    const float bias = b1[col];
#pragma unroll
    for (int v = 0; v < 8; ++v) {
      const int row = v + 8 * kh;
      float x = acc[ct][v] + bias;
      r_sh[row * RS + col] = x > 0.f ? x : 0.f;
    }
  }
  __syncthreads();

  // ---- GEMM2: wave wv computes logits columns [wv*16, wv*16+16), K = 256 ----
  v8f acc2 = vzero;
  const int col2 = wv * 16 + m;
  for (int kk = 0; kk < HID / 4; ++kk) {
    const int kb = kk * 4 + 2 * kh;
    v2f a;
    a[0] = r_sh[m * RS + kb];
    a[1] = r_sh[m * RS + kb + 1];
    v2f b;
    b[0] = W2[(size_t)kb * NBITS + col2];
    b[1] = W2[(size_t)(kb + 1) * NBITS + col2];
    acc2 = __builtin_amdgcn_wmma_f32_16x16x4_f32(
        false, a, false, b, (short)0, acc2, false, false);
  }
  const float bias2 = b2[col2];
#pragma unroll
  for (int v = 0; v < 8; ++v) {
    const int row = v + 8 * kh;
    l_sh[row * NBITS + col2] = acc2[v] + bias2;
  }
  __syncthreads();

  // ---- row softmax over 64 logits (one thread per row) ----
  if (t < 16) {
    float mx = -3.402823466e+38f;
    for (int c = 0; c < NBITS; ++c) mx = fmaxf(mx, l_sh[t * NBITS + c]);
    float s = 0.f;
    for (int c = 0; c < NBITS; ++c) {
      float e = expf(l_sh[t * NBITS + c] - mx);
      l_sh[t * NBITS + c] = e;
      s += e;
    }
    const float inv = 1.f / s;
    for (int c = 0; c < NBITS; ++c) l_sh[t * NBITS + c] *= inv;
  }
  __syncthreads();

  // ---- per-block column sums -> global fps accumulator ----
  if (t < NBITS) {
    float s = 0.f;
#pragma unroll
    for (int r = 0; r < 16; ++r) s += l_sh[r * NBITS + t];
    atomicAdd(&fps_acc[t], s);
  }
}

// ---------------------------------------------------------------------------
// Kernel 4: finalize.  fps = DEPTH * colsum;  out = relu(fps@W3+b3) @ W4 + b4
// ---------------------------------------------------------------------------
__global__ __launch_bounds__(256) void k_final(const float* __restrict__ fps_acc,
                                               const float* __restrict__ W3,
                                               const float* __restrict__ b3,
                                               const float* __restrict__ W4,
                                               const float* __restrict__ b4,
                                               float* __restrict__ out) {
  __shared__ float fps_sh[NBITS];
  __shared__ float red[256];
  const int t = threadIdx.x;
  if (t < NBITS) {
    const float f = fps_acc[t] * (float)DEPTH;   // DEPTH identical iterations
    fps_sh[t] = f;
    out[t] = f;                                   // d_out[0..63] = fps
  }
  __syncthreads();
  float a = b3[t];
  for (int k = 0; k < NBITS; ++k) a += fps_sh[k] * W3[(size_t)k * HID + t];
  a = a > 0.f ? a : 0.f;                          // relu
  red[t] = a * W4[t];                             // times W4[:,0]
  __syncthreads();
  for (int s = 128; s > 0; s >>= 1) {
    if (t < s) red[t] += red[t + s];
    __syncthreads();
  }
  if (t == 0) out[NBITS] = red[0] + b4[0];        // d_out[64] = out
}

// ---------------------------------------------------------------------------
extern "C" void kernel_launch(void* const* d_in, const int* in_sizes, int n_in,
                              void* d_out, int out_size, void* d_ws, size_t ws_size,
                              hipStream_t stream) {
  const float* n_feat = (const float*)d_in[0];
  const int*   src    = (const int*)d_in[1];
  const int*   dst    = (const int*)d_in[2];
  const float* W1     = (const float*)d_in[3];
  const float* b1     = (const float*)d_in[4];
  const float* W2     = (const float*)d_in[5];
  const float* b2     = (const float*)d_in[6];
  const float* W3     = (const float*)d_in[7];
  const float* b3     = (const float*)d_in[8];
  const float* W4     = (const float*)d_in[9];
  const float* b4     = (const float*)d_in[10];
  float* out = (float*)d_out;

  // workspace layout: h [N,DIM] f32, then fps accumulator [64] f32
  float* h       = (float*)d_ws;
  float* fps_acc = h + (size_t)N_NODES * DIM;

  const int n4 = (N_NODES * DIM) / 4;
  k_zero<<<(n4 + 255) / 256, 256, 0, stream>>>((float4*)h, fps_acc, n4);

  k_scatter<<<(N_EDGES * 32) / 256, 256, 0, stream>>>(
      (const float4*)n_feat, src, dst, h);

  k_node_mlp<<<N_NODES / 16, 128, 0, stream>>>(h, W1, b1, W2, b2, fps_acc);

  k_final<<<1, 256, 0, stream>>>(fps_acc, W3, b3, W4, b4, out);
}